// MoEGate_67242007986676
// MI455X (gfx1250) — compile-verified
//
#include <hip/hip_runtime.h>

typedef __attribute__((ext_vector_type(2))) float v2f;
typedef __attribute__((ext_vector_type(8))) float v8f;

#define N_TOKENS  16384
#define D_HIDDEN  4096
#define N_EXPERTS 64
#define TOP_K     8

#define TOK_PER_BLOCK 64
#define THREADS       128                 // 4 waves (wave32)
#define KC            32                  // K-chunk staged in LDS
#define NCHUNK        (D_HIDDEN / KC)     // 128
#define XS_STRIDE     36                  // 144B rows: 16B-aligned, conflict-free A reads
#define XS_BYTES      (TOK_PER_BLOCK * XS_STRIDE * 4)
#define WS_STRIDE     160                 // words per K-pair row; ≡32 mod 64 banks → conflict-free B reads
#define LG_STRIDE     65                  // conflict-free per-token row scans

__global__ __launch_bounds__(THREADS) void moe_gate_wmma_f32(
    const float* __restrict__ x, const float* __restrict__ Wg,
    int* __restrict__ topk_ids, float* __restrict__ topk_vals)
{
    __shared__ float xs[2][TOK_PER_BLOCK * XS_STRIDE];   // 2 x 9216 B (async-DMA filled)
    __shared__ float wsb[2][(KC / 2) * WS_STRIDE];       // 2 x 10240 B, K-pair interleaved W
    __shared__ float lg[TOK_PER_BLOCK * LG_STRIDE];      // 16640 B

    const int tid     = threadIdx.x;
    const int wave    = tid >> 5;
    const int lane    = tid & 31;
    const int l16     = lane & 15;
    const int hi      = lane >> 4;            // 0: K=0,1 half; 1: K=2,3 half
    const int tok0    = blockIdx.x * TOK_PER_BLOCK;
    const int waveTok = wave * 16;

    v8f c[4] = {};                            // 64 experts = 4 N-tiles of 16

    // ---- per-thread staging coordinates (loop-invariant) ----
    // x tile: 4 x 16B async copies per thread (64 rows x 32 floats)
    uint32_t lxo[4];                          // LDS byte address of each dst (buffer 0)
    uint64_t gxa[4];                          // global byte address of each src (chunk 0)
    #pragma unroll
    for (int r = 0; r < 4; ++r) {
        int i = r * THREADS + tid;
        int row = i >> 3, cf = i & 7;
        lxo[r] = (uint32_t)(uintptr_t)(&xs[0][row * XS_STRIDE + cf * 4]);
        gxa[r] = (uint64_t)(uintptr_t)(x + (size_t)(tok0 + row) * D_HIDDEN + cf * 4);
    }
    // W slice: 2 K-pair/expert-quad tasks per thread
    int wp[2], weq[2];
    #pragma unroll
    for (int q = 0; q < 2; ++q) { int i = q * THREADS + tid; wp[q] = i >> 4; weq[q] = i & 15; }

    float4 wa[2], wb[2];                      // W regs for next chunk (needs interleave transform)

    auto async_x = [&](int kc0, int buf) {    // gfx1250 async DMA: global -> LDS, no VGPR data
        #pragma unroll
        for (int r = 0; r < 4; ++r) {
            uint32_t l = lxo[r] + (uint32_t)buf * XS_BYTES;
            uint64_t g = gxa[r] + (uint64_t)kc0 * 4u;
            asm volatile("global_load_async_to_lds_b128 %0, %1, off"
                         :: "v"(l), "v"(g) : "memory");
        }
    };
    auto load_w = [&](int kc0) {
        #pragma unroll
        for (int q = 0; q < 2; ++q) {
            const float* wrow = Wg + (size_t)(kc0 + wp[q] * 2) * N_EXPERTS + weq[q] * 4;
            wa[q] = *reinterpret_cast<const float4*>(wrow);              // K row 2p
            wb[q] = *reinterpret_cast<const float4*>(wrow + N_EXPERTS);  // K row 2p+1
        }
    };
    auto store_w = [&](int buf) {
        #pragma unroll
        for (int q = 0; q < 2; ++q) {
            float* dst = &wsb[buf][wp[q] * WS_STRIDE + weq[q] * 8];
            v2f p0; p0.x = wa[q].x; p0.y = wb[q].x;
            v2f p1; p1.x = wa[q].y; p1.y = wb[q].y;
            v2f p2; p2.x = wa[q].z; p2.y = wb[q].z;
            v2f p3; p3.x = wa[q].w; p3.y = wb[q].w;
            *reinterpret_cast<v2f*>(dst + 0) = p0;   // {W[2p][e], W[2p+1][e]} contiguous
            *reinterpret_cast<v2f*>(dst + 2) = p1;
            *reinterpret_cast<v2f*>(dst + 4) = p2;
            *reinterpret_cast<v2f*>(dst + 6) = p3;
        }
    };

    // ---- prologue: stage chunk 0 ----
    async_x(0, 0);
    load_w(0);
    store_w(0);
    asm volatile("s_wait_asynccnt 0x0" ::: "memory");
    __syncthreads();

    int buf = 0;
    for (int ch = 0; ch < NCHUNK; ++ch) {
        const bool has_next = (ch + 1) < NCHUNK;
        if (has_next) {
            async_x((ch + 1) * KC, buf ^ 1);  // DMA next x tile while WMMAs run
            load_w((ch + 1) * KC);
        }

        #pragma unroll
        for (int s = 0; s < KC / 4; ++s) {
            v2f a = *reinterpret_cast<const v2f*>(
                &xs[buf][(waveTok + l16) * XS_STRIDE + s * 4 + hi * 2]);
            const float* bbase = &wsb[buf][(s * 2 + hi) * WS_STRIDE + l16 * 2];
            #pragma unroll
            for (int t = 0; t < 4; ++t) {
                v2f b = *reinterpret_cast<const v2f*>(bbase + t * 32);  // single ds_load_b64
                c[t] = __builtin_amdgcn_wmma_f32_16x16x4_f32(
                    false, a, false, b, (short)0, c[t], false, false);
            }
        }

        if (has_next) store_w(buf ^ 1);
        asm volatile("s_wait_asynccnt 0x0" ::: "memory");  // publish DMA'd tile
        __syncthreads();                                   // one barrier per chunk
        buf ^= 1;
    }

    // Spill logits: C layout — VGPR j holds M=j (lanes 0-15) / M=j+8 (lanes 16-31)
    #pragma unroll
    for (int t = 0; t < 4; ++t)
        #pragma unroll
        for (int j = 0; j < 8; ++j)
            lg[(waveTok + hi * 8 + j) * LG_STRIDE + t * 16 + l16] = c[t][j];
    __syncthreads();

    // One thread per token: softmax stats + 8 argmax passes (strict '>' keeps the
    // lowest index on ties, matching lax.top_k's descending sorted order).
    if (tid < TOK_PER_BLOCK) {
        float* row = &lg[tid * LG_STRIDE];
        float m = row[0];
        for (int e = 1; e < N_EXPERTS; ++e) m = fmaxf(m, row[e]);
        float sum = 0.0f;
        for (int e = 0; e < N_EXPERTS; ++e) sum += __expf(row[e] - m);
        const float inv = 1.0f / sum;
        const int token = tok0 + tid;
        for (int k = 0; k < TOP_K; ++k) {
            float best = -__builtin_inff();
            int bi = 0;
            for (int e = 0; e < N_EXPERTS; ++e) {
                float v = row[e];
                if (v > best) { best = v; bi = e; }
            }
            row[bi] = -__builtin_inff();
            topk_ids[token * TOP_K + k]  = bi;
            topk_vals[token * TOP_K + k] = __expf(best - m) * inv;
        }
    }
}

extern "C" void kernel_launch(void* const* d_in, const int* in_sizes, int n_in,
                              void* d_out, int out_size, void* d_ws, size_t ws_size,
                              hipStream_t stream) {
    (void)in_sizes; (void)n_in; (void)out_size; (void)d_ws; (void)ws_size;
    const float* x  = (const float*)d_in[0];
    const float* Wg = (const float*)d_in[1];
    int*   ids  = (int*)d_out;                               // topk_ids first
    float* vals = (float*)d_out + (size_t)N_TOKENS * TOP_K;  // then topk_vals
    dim3 grid(N_TOKENS / TOK_PER_BLOCK), block(THREADS);
    hipLaunchKernelGGL(moe_gate_wmma_f32, grid, block, 0, stream,
                       x, Wg, ids, vals);
}